// Loss_SSIM_39307540693206
// MI455X (gfx1250) — compile-verified
//
#include <hip/hip_runtime.h>

// MI455X / gfx1250, wave32. SSIM via fp32 WMMA band-matrix convolutions,
// with TDM (tensor_load_to_lds) staging of input tiles for interior tiles.

typedef float v2f __attribute__((ext_vector_type(2)));
typedef float v8f __attribute__((ext_vector_type(8)));
typedef unsigned int v4u __attribute__((ext_vector_type(4)));
typedef int v4i __attribute__((ext_vector_type(4)));
typedef int v8i __attribute__((ext_vector_type(8)));

// 11-tap Gaussian, sigma=1.5, normalized (matches reference _gaussian_window)
__constant__ float c_g[11] = {
    0.00102838f, 0.00759876f, 0.03600077f, 0.10936070f, 0.21300553f,
    0.26601172f,
    0.21300553f, 0.10936070f, 0.03600077f, 0.00759876f, 0.00102838f
};

__device__ __forceinline__ float gband(int d) {
    return ((unsigned)d <= 10u) ? c_g[d] : 0.0f;
}

// D = A(16x4) * B(4x16) + C, full fp32
__device__ __forceinline__ v8f wmma_f32(v2f a, v2f b, v8f c) {
    return __builtin_amdgcn_wmma_f32_16x16x4_f32(
        /*neg_a=*/false, a, /*neg_b=*/false, b,
        /*c_mod=*/(short)0, c, /*reuse_a=*/false, /*reuse_b=*/false);
}

#define PLANE_STRIDE (512 * 512)
#define N_PLANES     48            // 16 * 3
#define TILES_PER_PLANE 1024
#define N_TILES      (N_PLANES * TILES_PER_PLANE)   // 49152
#define WAVES_PER_BLK 4
#define N_BLOCKS     (N_TILES / WAVES_PER_BLK)      // 12288

__device__ __forceinline__ unsigned rfl(unsigned v) {
    return (unsigned)__builtin_amdgcn_readfirstlane((int)v);
}

// One TDM DMA: load a 28-row x 32-col fp32 tile (row stride 512 floats in
// memory, contiguous 32-float rows in LDS) starting at gptr. Elements past
// validX cols / validY rows are hardware-zeroed (tensor OOB). TENSORcnt++.
__device__ __forceinline__ void tdm_load_tile(const float* gptr, float* lptr,
                                              int validX, int validY)
{
    unsigned long long ga = (unsigned long long)(uintptr_t)gptr;  // 57-bit byte addr
    unsigned lds = (unsigned)(uintptr_t)lptr;                     // LDS byte offset

    // ---- D# group 0 (128b): count=1 | lds_addr | global_addr | type=2 ----
    v4u g0 = {
        rfl(1u),                                              // count=1, user mode
        rfl(lds),                                             // bits 63:32 lds_addr
        rfl((unsigned)ga),                                    // bits 95:64 ga[31:0]
        rfl(((unsigned)(ga >> 32) & 0x01FFFFFFu) | (2u << 30))// ga[56:32] | type=2
    };

    // ---- D# group 1 (256b) ----
    const unsigned d0 = (unsigned)validX;     // tensor_dim0 (valid cols from tile start)
    const unsigned d1 = (unsigned)validY;     // tensor_dim1 (valid rows from tile start)
    v8i g1 = {
        (int)rfl(2u << 16),                       // data_size=4B; no mask/pad/iterate
        (int)rfl((d0 & 0xFFFFu) << 16),           // atomic_barrier=0 | tensor_dim0 lo16
        (int)rfl((d0 >> 16) | ((d1 & 0xFFFFu) << 16)), // tensor_dim0 hi | tensor_dim1 lo
        (int)rfl((d1 >> 16) | (32u << 16)),       // tensor_dim1 hi | tile_dim0=32
        (int)rfl(28u),                            // tile_dim1=28 | tile_dim2=0
        (int)rfl(512u),                           // tensor_dim0_stride = 512 (lo32)
        0,                                        // stride hi | tensor_dim1_stride lo (2D: unused)
        0                                         // tensor_dim1_stride hi (unused)
    };
    v4i gz4 = {0, 0, 0, 0};                       // groups 2/3 unused (2D tile)
    v8i gz8 = {0, 0, 0, 0, 0, 0, 0, 0};           // extra group (6-arg builtin form)

    __builtin_amdgcn_tensor_load_to_lds(g0, g1, gz4, gz4, gz8, 0);
}

__global__ __launch_bounds__(128)
void ssim_wmma_kernel(const float* __restrict__ img1,
                      const float* __restrict__ img2,
                      float* __restrict__ partials)
{
    // Per-wave private LDS slices (no cross-wave sharing -> no barriers in hot path;
    // LDS ops are in-order within a wave on CDNA5).
    __shared__ float sIn[WAVES_PER_BLK][2][28 * 32]; // [wave][x|y][row 0..27][col 0..31]
    __shared__ float sVt[WAVES_PER_BLK][16 * 32];    // stage-1 result V
    __shared__ float sRed[WAVES_PER_BLK];

    const int lane = threadIdx.x & 31;
    const int wave = threadIdx.x >> 5;
    const bool hi  = lane >= 16;        // K-half selector per A/B layouts
    const int  l16 = lane & 15;

    const int tile  = blockIdx.x * WAVES_PER_BLK + wave;   // 0..49151
    const int plane = tile >> 10;                          // 0..47  (n*3+c)
    const int trem  = tile & 1023;
    const int ty    = trem >> 5;
    const int tx    = trem & 31;
    const int pOff  = plane * PLANE_STRIDE;
    const int gy0   = ty * 16 - 5;
    const int gx0   = tx * 16 - 5;

    float* inX = sIn[wave][0];
    float* inY = sIn[wave][1];
    float* vt  = sVt[wave];

    // ---- Stage inputs into LDS ----------------------------------------------
    // The band matrices have zero taps for staged rows 26..27 and cols 26..31,
    // so those cells only need to be FINITE (any real pixel is fine). True
    // zeros are required only where the conv's zero padding applies:
    //   * right/bottom image edge: TDM tensor-OOB returns zeros automatically
    //   * top/left image edge (gy0<0 or gx0<0): not expressible as a TDM tile,
    //     so those (wave-uniform) border tiles use the scalar zero-padded path.
    if (tx > 0 && ty > 0) {
        // Interior: one TDM DMA per input plane, wave-level (EXEC ignored).
        const float* gx_ = img1 + pOff + gy0 * 512 + gx0;
        const float* gy_ = img2 + pOff + gy0 * 512 + gx0;
        tdm_load_tile(gx_, inX, 512 - gx0, 512 - gy0);
        tdm_load_tile(gy_, inY, 512 - gx0, 512 - gy0);
        __builtin_amdgcn_s_wait_tensorcnt(0);
        asm volatile("" ::: "memory");   // keep LDS reads below the wait
    } else {
        for (int idx = lane; idx < 28 * 32; idx += 32) {
            const int r = idx >> 5, c = idx & 31;
            float vx = 0.0f, vy = 0.0f;
            if (r < 26 && c < 26) {
                const int gy = gy0 + r, gx = gx0 + c;
                if ((unsigned)gy < 512u && (unsigned)gx < 512u) {
                    const int off = pOff + gy * 512 + gx;
                    vx = img1[off];
                    vy = img2[off];
                }
            }
            inX[idx] = vx;
            inY[idx] = vy;
        }
    }

    // ---- Constant Gaussian band fragments. A-layout (M=l16) and B-layout
    // (N=l16) mirror each other, so one fragment serves both stages:
    //   stage1 A[i][r] = g[r-i],  stage2 B[r][j] = g[r-j],  value = g[r - l16].
    v2f Wc[7];
#pragma unroll
    for (int kk = 0; kk < 7; ++kk) {
        const int r = kk * 4 + (hi ? 2 : 0);
        Wc[kk].x = gband(r - l16);
        Wc[kk].y = gband(r + 1 - l16);
    }

    v8f outp[5];

    // Planes: 0=x, 1=y, 2=x*x, 3=y*y, 4=x*y (product planes formed per-lane
    // in the elementwise B layout -> no extra LDS storage).
#pragma unroll
    for (int p = 0; p < 5; ++p) {
        v8f d0 = {0.f,0.f,0.f,0.f,0.f,0.f,0.f,0.f};
        v8f d1 = {0.f,0.f,0.f,0.f,0.f,0.f,0.f,0.f};

        // ---- Stage 1 (vertical): V(16x26) = Gband(16x28) * In(28x26), 2 N-tiles ----
#pragma unroll
        for (int kk = 0; kk < 7; ++kk) {
            const int r0 = kk * 4 + (hi ? 2 : 0);
            const int c0 = l16;
            const int c1 = 16 + l16;
            v2f b0, b1;
            if (p == 0) {
                b0.x = inX[r0 * 32 + c0]; b0.y = inX[(r0 + 1) * 32 + c0];
                b1.x = inX[r0 * 32 + c1]; b1.y = inX[(r0 + 1) * 32 + c1];
            } else if (p == 1) {
                b0.x = inY[r0 * 32 + c0]; b0.y = inY[(r0 + 1) * 32 + c0];
                b1.x = inY[r0 * 32 + c1]; b1.y = inY[(r0 + 1) * 32 + c1];
            } else if (p == 2) {
                float a0 = inX[r0 * 32 + c0], a1 = inX[(r0 + 1) * 32 + c0];
                float a2 = inX[r0 * 32 + c1], a3 = inX[(r0 + 1) * 32 + c1];
                b0.x = a0 * a0; b0.y = a1 * a1; b1.x = a2 * a2; b1.y = a3 * a3;
            } else if (p == 3) {
                float a0 = inY[r0 * 32 + c0], a1 = inY[(r0 + 1) * 32 + c0];
                float a2 = inY[r0 * 32 + c1], a3 = inY[(r0 + 1) * 32 + c1];
                b0.x = a0 * a0; b0.y = a1 * a1; b1.x = a2 * a2; b1.y = a3 * a3;
            } else {
                float x0 = inX[r0 * 32 + c0], x1 = inX[(r0 + 1) * 32 + c0];
                float x2 = inX[r0 * 32 + c1], x3 = inX[(r0 + 1) * 32 + c1];
                float y0 = inY[r0 * 32 + c0], y1 = inY[(r0 + 1) * 32 + c0];
                float y2 = inY[r0 * 32 + c1], y3 = inY[(r0 + 1) * 32 + c1];
                b0.x = x0 * y0; b0.y = x1 * y1; b1.x = x2 * y2; b1.y = x3 * y3;
            }
            d0 = wmma_f32(Wc[kk], b0, d0);
            d1 = wmma_f32(Wc[kk], b1, d1);
        }

        // ---- D-layout -> LDS (row-major V[16][32]) ----
        const int rb = hi ? 8 : 0;
#pragma unroll
        for (int r = 0; r < 8; ++r) {
            vt[(r + rb) * 32 + l16]      = d0[r];
            vt[(r + rb) * 32 + 16 + l16] = d1[r];
        }

        // ---- Stage 2 (horizontal): O(16x16) = V(16x28) * Gband(28x16) ----
        v8f o = {0.f,0.f,0.f,0.f,0.f,0.f,0.f,0.f};
#pragma unroll
        for (int kk = 0; kk < 7; ++kk) {
            const int k0 = kk * 4 + (hi ? 2 : 0);
            v2f a;
            a.x = vt[l16 * 32 + k0];
            a.y = vt[l16 * 32 + k0 + 1];
            o = wmma_f32(a, Wc[kk], o);
        }
        outp[p] = o;
    }

    // ---- SSIM map + wave/block reduction (deterministic, no atomics) ----
    const float C1 = 0.0001f, C2 = 0.0009f;
    float s = 0.0f;
#pragma unroll
    for (int r = 0; r < 8; ++r) {
        const float m1 = outp[0][r], m2 = outp[1][r];
        const float exx = outp[2][r], eyy = outp[3][r], exy = outp[4][r];
        const float m1s = m1 * m1, m2s = m2 * m2, m12 = m1 * m2;
        const float s1 = exx - m1s, s2 = eyy - m2s, s12 = exy - m12;
        const float num = (2.0f * m12 + C1) * (2.0f * s12 + C2);
        const float den = (m1s + m2s + C1) * (s1 + s2 + C2);
        s += num / den;
    }
#pragma unroll
    for (int off = 16; off > 0; off >>= 1)
        s += __shfl_xor(s, off, 32);
    if (lane == 0) sRed[wave] = s;
    __syncthreads();
    if (threadIdx.x == 0)
        partials[blockIdx.x] = (sRed[0] + sRed[1]) + (sRed[2] + sRed[3]);
}

__global__ __launch_bounds__(256)
void ssim_reduce_kernel(const float* __restrict__ partials, int n,
                        float* __restrict__ out)
{
    __shared__ float sm[256];
    float s = 0.0f;
    for (int i = threadIdx.x; i < n; i += 256) s += partials[i];
    sm[threadIdx.x] = s;
    __syncthreads();
    for (int w = 128; w > 0; w >>= 1) {
        if ((int)threadIdx.x < w) sm[threadIdx.x] += sm[threadIdx.x + w];
        __syncthreads();
    }
    if (threadIdx.x == 0)
        out[0] = sm[0] * (1.0f / 12582912.0f);   // / (16*3*512*512)
}

extern "C" void kernel_launch(void* const* d_in, const int* in_sizes, int n_in,
                              void* d_out, int out_size, void* d_ws, size_t ws_size,
                              hipStream_t stream)
{
    (void)in_sizes; (void)n_in; (void)out_size; (void)ws_size;
    const float* img1 = (const float*)d_in[0];
    const float* img2 = (const float*)d_in[1];
    float* out      = (float*)d_out;
    float* partials = (float*)d_ws;   // N_BLOCKS floats = 48 KB scratch

    ssim_wmma_kernel<<<N_BLOCKS, 128, 0, stream>>>(img1, img2, partials);
    ssim_reduce_kernel<<<1, 256, 0, stream>>>(partials, N_BLOCKS, out);
}